// AutoregressiveLSTM_49735721288032
// MI455X (gfx1250) — compile-verified
//
#include <hip/hip_runtime.h>
#include <math.h>

// ---------------------------------------------------------------------------
// AutoregressiveLSTM for MI455X (gfx1250, wave32).
//   SEQ=512, PRED=96, H=1024, F=32, B=1, 2 layers.
// Recurrent GEMVs: persistent 32-WG kernels, grid barrier per timestep,
// weights served from L2 (transposed for coalescing).
// Layer-1 input projection (512x1024 @ 1024x4096): fp32 WMMA 16x16x4,
// N-blocked 1x4 per wave so each A fragment feeds 4 WMMAs.
// ---------------------------------------------------------------------------

#define SEQ 512
#define PRED 96
#define HID 1024
#define FEAT 32
#define G4H 4096
#define NWG_LSTM 32

typedef __attribute__((ext_vector_type(2))) float v2f;
typedef __attribute__((ext_vector_type(8))) float v8f;

// ------------------------- grid barrier (monotone counter) -----------------
__device__ __forceinline__ void grid_barrier(unsigned* cnt, unsigned target) {
  __syncthreads();
  if (threadIdx.x == 0) {
    __hip_atomic_fetch_add(cnt, 1u, __ATOMIC_RELEASE, __HIP_MEMORY_SCOPE_AGENT);
    while (__hip_atomic_load(cnt, __ATOMIC_ACQUIRE, __HIP_MEMORY_SCOPE_AGENT) < target) {
      __builtin_amdgcn_s_sleep(1);
    }
  }
  __syncthreads();
  __threadfence();
}

// ------------------------- init: biases, stats, state, counters ------------
__global__ void init_all(const float* __restrict__ x,
                         const float* __restrict__ bih0, const float* __restrict__ bhh0,
                         const float* __restrict__ bih1, const float* __restrict__ bhh1,
                         float* __restrict__ buf, float* __restrict__ bias0,
                         float* __restrict__ bias1,
                         float* __restrict__ h0, float* __restrict__ c0,
                         float* __restrict__ h1, float* __restrict__ c1,
                         float* __restrict__ mean, float* __restrict__ stdv,
                         unsigned* __restrict__ counters) {
  const int t = threadIdx.x;  // 1024 threads, one block
  for (int idx = t; idx < SEQ * FEAT; idx += 1024) buf[idx] = x[idx];
  for (int idx = t; idx < G4H; idx += 1024) {
    bias0[idx] = bih0[idx] + bhh0[idx];
    bias1[idx] = bih1[idx] + bhh1[idx];
  }
  h0[t] = 0.f; c0[t] = 0.f; h1[t] = 0.f; c1[t] = 0.f;
  if (t < 256) counters[t] = 0u;
  if (t < FEAT) {
    float m = 0.f;
    for (int s = 0; s < SEQ; ++s) m += x[s * FEAT + t];
    m *= (1.f / SEQ);
    float v = 0.f;
    for (int s = 0; s < SEQ; ++s) { float d = x[s * FEAT + t] - m; v += d * d; }
    v *= (1.f / SEQ);
    mean[t] = m;
    stdv[t] = sqrtf(v + 1e-5f);
  }
}

// ------------------------- transpose [R][C] -> [C][R] -----------------------
__global__ void transpose_k(const float* __restrict__ src, float* __restrict__ dst,
                            int R, int C) {
  const int n = R * C;
  for (int idx = blockIdx.x * blockDim.x + threadIdx.x; idx < n;
       idx += gridDim.x * blockDim.x) {
    const int r = idx / C, c = idx - r * C;
    dst[c * R + r] = src[idx];
  }
}

// ------------------------- recurrent LSTM layer (persistent) ----------------
// 32 WGs x 256 threads. WG owns 32 hidden units (all four gates i,f,g,o).
// thread t: out_id = t&127 = gate*32+uu ; khalf = t>>7 splits the K=1024 dot.
template <int LAYER>
__global__ __launch_bounds__(256)
void lstm_seq(const float* __restrict__ WT,     // [1024][4096] transposed W_hh
              const float* __restrict__ WTih0,  // [32][4096] transposed W_ih0 (L0)
              const float* __restrict__ xin,    // L0: [*][32] window base; L1: [512][4096]
              const float* __restrict__ bias,   // [4096] combined b_ih+b_hh
              float* __restrict__ hG,           // [1024] carried state
              float* __restrict__ cG,           // [1024] carried state
              float* __restrict__ yout,         // L0: [512][1024]; L1 unused
              unsigned* __restrict__ counter) {
  __shared__ float hs[HID];
  __shared__ float red[256];
  __shared__ float gv[128];
  __shared__ float cs[32];

  const int t = threadIdx.x;
  const int jbase = blockIdx.x * 32;
  const int out_id = t & 127;
  const int khalf = t >> 7;
  const int gate = out_id >> 5;
  const int uu = out_id & 31;
  const int g = gate * HID + jbase + uu;

  if (t < 32) cs[t] = cG[jbase + t];

  for (int ts = 0; ts < SEQ; ++ts) {
    __syncthreads();
#pragma unroll
    for (int q = 0; q < 4; ++q) hs[t + q * 256] = hG[t + q * 256];
    __syncthreads();

    float acc = 0.f;
    const float* wcol = WT + (size_t)khalf * 512 * G4H + g;
    const float* hptr = hs + khalf * 512;
#pragma unroll 8
    for (int k = 0; k < 512; ++k) acc += hptr[k] * wcol[(size_t)k * G4H];

    if (khalf == 0) {
      if (LAYER == 0) {
        const float* xr = xin + ts * FEAT;
        const float* wi = WTih0 + g;
#pragma unroll
        for (int q = 0; q < FEAT; ++q) acc += xr[q] * wi[q * G4H];
        acc += bias[g];
      } else {
        acc += xin[(size_t)ts * G4H + g] + bias[g];
      }
    }
    red[t] = acc;
    __syncthreads();
    if (t < 128) gv[t] = red[t] + red[t + 128];
    __syncthreads();
    if (t < 32) {
      const float iv = gv[t], fv = gv[32 + t], gg = gv[64 + t], ov = gv[96 + t];
      const float is = 1.f / (1.f + __expf(-iv));
      const float fs = 1.f / (1.f + __expf(-fv));
      const float os = 1.f / (1.f + __expf(-ov));
      const float c = fs * cs[t] + is * tanhf(gg);
      const float h = os * tanhf(c);
      cs[t] = c;
      hG[jbase + t] = h;
      if (LAYER == 0) yout[(size_t)ts * HID + jbase + t] = h;
    }
    grid_barrier(counter, (unsigned)(ts + 1) * NWG_LSTM);
  }
  if (t < 32) cG[jbase + t] = cs[t];
}

// ------------------------- x_proj1 GEMM: fp32 WMMA 16x16x4 ------------------
// C[512][4096] = A[512][1024] * B[4096][1024]^T  (B row n = W_ih1 row g, K-major)
// One wave per 16x64 strip (4 N-tiles): each A fragment load feeds 4 WMMAs.
__global__ __launch_bounds__(256)
void xp1_gemm(const float* __restrict__ A, const float* __restrict__ Bm,
              float* __restrict__ C) {
  const int wave = (blockIdx.x * blockDim.x + threadIdx.x) >> 5;  // 0..2047
  const int lane = threadIdx.x & 31;
  const int mt = wave >> 6;    // 0..31  (M tiles of 512/16)
  const int ns = wave & 63;    // 0..63  (N strips of 64 cols)
  const int half = lane >> 4;  // K-pair select per ISA 16x4 fp32 A layout
  const int l15 = lane & 15;

  const float* arow = A + (size_t)(mt * 16 + l15) * HID + 2 * half;
  const float* brow = Bm + (size_t)(ns * 64 + l15) * HID + 2 * half;

  v8f acc0 = {}, acc1 = {}, acc2 = {}, acc3 = {};
#pragma unroll 2
  for (int k = 0; k < HID; k += 4) {
    v2f a  = *(const v2f*)(arow + k);
    v2f b0 = *(const v2f*)(brow + k);
    v2f b1 = *(const v2f*)(brow + 16 * HID + k);
    v2f b2 = *(const v2f*)(brow + 32 * HID + k);
    v2f b3 = *(const v2f*)(brow + 48 * HID + k);
    acc0 = __builtin_amdgcn_wmma_f32_16x16x4_f32(false, a, false, b0, (short)0,
                                                 acc0, false, false);
    acc1 = __builtin_amdgcn_wmma_f32_16x16x4_f32(false, a, false, b1, (short)0,
                                                 acc1, false, false);
    acc2 = __builtin_amdgcn_wmma_f32_16x16x4_f32(false, a, false, b2, (short)0,
                                                 acc2, false, false);
    acc3 = __builtin_amdgcn_wmma_f32_16x16x4_f32(false, a, false, b3, (short)0,
                                                 acc3, false, false);
  }
  // C layout per tile: VGPR v -> row m = v + 8*half, col n = l15.
  float* cbase = C + (size_t)(mt * 16) * G4H + ns * 64 + l15;
#pragma unroll
  for (int v = 0; v < 8; ++v) {
    float* crow = cbase + (size_t)(v + 8 * half) * G4H;
    crow[0]  = acc0[v];
    crow[16] = acc1[v];
    crow[32] = acc2[v];
    crow[48] = acc3[v];
  }
}

// ------------------------- head: fc + renorm + append -----------------------
__global__ void head_fc(const float* __restrict__ h1, const float* __restrict__ fcw,
                        const float* __restrict__ fcb, const float* __restrict__ mean,
                        const float* __restrict__ stdv, float* __restrict__ out,
                        float* __restrict__ bufrow) {
  __shared__ float red[256];
  const int t = threadIdx.x;
  const int f = t >> 3, part = t & 7;
  const float* wr = fcw + f * HID + part * 128;
  const float* hp = h1 + part * 128;
  float acc = 0.f;
#pragma unroll 8
  for (int k = 0; k < 128; ++k) acc += hp[k] * wr[k];
  red[t] = acc;
  __syncthreads();
  if (part == 0) {
    float v = fcb[f];
#pragma unroll
    for (int q = 0; q < 8; ++q) v += red[f * 8 + q];
    out[f] = v;
    bufrow[f] = (v - mean[f]) / stdv[f];
  }
}

// ---------------------------------------------------------------------------
extern "C" void kernel_launch(void* const* d_in, const int* in_sizes, int n_in,
                              void* d_out, int out_size, void* d_ws, size_t ws_size,
                              hipStream_t stream) {
  const float* x     = (const float*)d_in[0];
  const float* W_ih0 = (const float*)d_in[1];
  const float* W_hh0 = (const float*)d_in[2];
  const float* b_ih0 = (const float*)d_in[3];
  const float* b_hh0 = (const float*)d_in[4];
  const float* W_ih1 = (const float*)d_in[5];
  const float* W_hh1 = (const float*)d_in[6];
  const float* b_ih1 = (const float*)d_in[7];
  const float* b_hh1 = (const float*)d_in[8];
  const float* fc_w  = (const float*)d_in[9];
  const float* fc_b  = (const float*)d_in[10];
  float* out = (float*)d_out;

  float* ws = (float*)d_ws;
  size_t o = 0;
  float* WT0   = ws + o; o += (size_t)G4H * HID;   // transposed W_hh0 [k][g]
  float* WT1   = ws + o; o += (size_t)G4H * HID;   // transposed W_hh1 [k][g]
  float* WTih0 = ws + o; o += (size_t)FEAT * G4H;  // transposed W_ih0 [k][g]
  float* buf   = ws + o; o += (size_t)(SEQ + PRED) * FEAT;
  float* y0    = ws + o; o += (size_t)SEQ * HID;
  float* xp1   = ws + o; o += (size_t)SEQ * G4H;
  float* bias0 = ws + o; o += G4H;
  float* bias1 = ws + o; o += G4H;
  float* h0    = ws + o; o += HID;
  float* c0    = ws + o; o += HID;
  float* h1    = ws + o; o += HID;
  float* c1    = ws + o; o += HID;
  float* mean  = ws + o; o += FEAT;
  float* stdv  = ws + o; o += FEAT;
  unsigned* counters = (unsigned*)(ws + o); o += 256;
  (void)in_sizes; (void)n_in; (void)out_size; (void)ws_size;

  init_all<<<1, 1024, 0, stream>>>(x, b_ih0, b_hh0, b_ih1, b_hh1, buf, bias0,
                                   bias1, h0, c0, h1, c1, mean, stdv, counters);
  transpose_k<<<512, 256, 0, stream>>>(W_hh0, WT0, G4H, HID);
  transpose_k<<<512, 256, 0, stream>>>(W_hh1, WT1, G4H, HID);
  transpose_k<<<64, 256, 0, stream>>>(W_ih0, WTih0, G4H, FEAT);

  for (int i = 0; i < PRED; ++i) {
    lstm_seq<0><<<NWG_LSTM, 256, 0, stream>>>(WT0, WTih0, buf + i * FEAT, bias0,
                                              h0, c0, y0, counters + i * 2);
    xp1_gemm<<<256, 256, 0, stream>>>(y0, W_ih1, xp1);
    lstm_seq<1><<<NWG_LSTM, 256, 0, stream>>>(WT1, nullptr, xp1, bias1, h1, c1,
                                              nullptr, counters + i * 2 + 1);
    head_fc<<<1, 256, 0, stream>>>(h1, fc_w, fc_b, mean, stdv, out + i * FEAT,
                                   buf + (size_t)(SEQ + i) * FEAT);
  }
}